// UnsupervisedGCN_49555332661645
// MI455X (gfx1250) — compile-verified
//
#include <hip/hip_runtime.h>

typedef __attribute__((ext_vector_type(2))) float v2f;
typedef __attribute__((ext_vector_type(8))) float v8f;

#define HID 64

// ---------------- zero fill ----------------
__global__ void zero_f32(float* __restrict__ p, int n) {
  int i = blockIdx.x * blockDim.x + threadIdx.x;
  if (i < n) p[i] = 0.0f;
}

// ---------------- degree histograms ----------------
__global__ void degree_kernel(const int* __restrict__ src, const int* __restrict__ dst,
                              float* __restrict__ deg_out, float* __restrict__ deg_in,
                              int E) {
  int e = blockIdx.x * blockDim.x + threadIdx.x;
  if (e < E) {
    unsafeAtomicAdd(&deg_out[src[e]], 1.0f);  // global_atomic_add_f32, no return
    unsafeAtomicAdd(&deg_in[dst[e]], 1.0f);
  }
}

// ---------------- inv-sqrt in place ----------------
__global__ void rsqrt_kernel(float* __restrict__ d, int n) {
  int i = blockIdx.x * blockDim.x + threadIdx.x;
  if (i < n) d[i] = rsqrtf(fmaxf(d[i], 1.0f));
}

// ---------------- edge scatter: agg[dst] += coef * feature[src] ----------------
// 16 lanes per edge, float4 per lane -> coalesced 256B feature row per edge.
__global__ void scatter_kernel(const float* __restrict__ feat,
                               const int* __restrict__ src, const int* __restrict__ dst,
                               const float* __restrict__ inv_out,
                               const float* __restrict__ inv_in,
                               float* __restrict__ agg, int E) {
  long long tid = (long long)blockIdx.x * blockDim.x + threadIdx.x;
  int e = (int)(tid >> 4);
  if (e >= E) return;
  int c = ((int)tid & 15) * 4;
  int s = src[e];
  int d = dst[e];
  float coef = inv_out[s] * inv_in[d];
  const float4 v = *reinterpret_cast<const float4*>(feat + (long long)s * HID + c);
  float* base = agg + (long long)d * HID + c;
  unsafeAtomicAdd(base + 0, v.x * coef);
  unsafeAtomicAdd(base + 1, v.y * coef);
  unsafeAtomicAdd(base + 2, v.z * coef);
  unsafeAtomicAdd(base + 3, v.w * coef);
}

// ---------------- GEMM: out = agg @ W + b, in place (out == agg) ----------------
// One wave -> 16 rows x 64 cols. K=64 in steps of 4 via V_WMMA_F32_16X16X4_F32.
__global__ void gemm_wmma_kernel(float* __restrict__ agg_out,  // [N,64] in, overwritten
                                 const float* __restrict__ W,  // [64,64] row-major
                                 const float* __restrict__ bias, // [64]
                                 int n_tiles) {
  int wave = threadIdx.x >> 5;
  int lane = threadIdx.x & 31;
  int tile = blockIdx.x * (blockDim.x >> 5) + wave;
  if (tile >= n_tiles) return;              // wave-uniform: EXEC stays all-1s
  int row0 = tile * 16;
  int lo = lane & 15;
  int hi = lane >> 4;                       // 0 or 1

  v8f acc0 = {}, acc1 = {}, acc2 = {}, acc3 = {};
  const float* arow = agg_out + (long long)(row0 + lo) * HID;

  #pragma unroll
  for (int kk = 0; kk < HID; kk += 4) {
    int k = kk + 2 * hi;                    // this lane's K pair
    v2f a;
    a.x = arow[k];
    a.y = arow[k + 1];

    const float* w0 = W + (long long)k * HID;        // row k
    const float* w1 = W + (long long)(k + 1) * HID;  // row k+1
    v2f b0, b1, b2, b3;
    b0.x = w0[lo];       b0.y = w1[lo];
    b1.x = w0[16 + lo];  b1.y = w1[16 + lo];
    b2.x = w0[32 + lo];  b2.y = w1[32 + lo];
    b3.x = w0[48 + lo];  b3.y = w1[48 + lo];

    acc0 = __builtin_amdgcn_wmma_f32_16x16x4_f32(false, a, false, b0, (short)0, acc0, false, false);
    acc1 = __builtin_amdgcn_wmma_f32_16x16x4_f32(false, a, false, b1, (short)0, acc1, false, false);
    acc2 = __builtin_amdgcn_wmma_f32_16x16x4_f32(false, a, false, b2, (short)0, acc2, false, false);
    acc3 = __builtin_amdgcn_wmma_f32_16x16x4_f32(false, a, false, b3, (short)0, acc3, false, false);
  }

  float bb0 = bias[lo];
  float bb1 = bias[16 + lo];
  float bb2 = bias[32 + lo];
  float bb3 = bias[48 + lo];

  #pragma unroll
  for (int r = 0; r < 8; ++r) {
    long long m = row0 + r + 8 * hi;        // C/D layout: VGPR r -> row r (+8 for hi lanes)
    float* orow = agg_out + m * HID;
    orow[lo]      = acc0[r] + bb0;
    orow[16 + lo] = acc1[r] + bb1;
    orow[32 + lo] = acc2[r] + bb2;
    orow[48 + lo] = acc3[r] + bb3;
  }
}

extern "C" void kernel_launch(void* const* d_in, const int* in_sizes, int n_in,
                              void* d_out, int out_size, void* d_ws, size_t ws_size,
                              hipStream_t stream) {
  const float* feature = (const float*)d_in[0];   // [N,64]
  const int*   src     = (const int*)d_in[1];     // [E]
  const int*   dst     = (const int*)d_in[2];     // [E]
  // d_in[3] = W1, d_in[4] = b1 : output does not depend on them (layer-1 result discarded)
  const float* W2      = (const float*)d_in[5];   // [64,64]
  const float* b2      = (const float*)d_in[6];   // [64]

  const int N = in_sizes[0] / HID;                // 100000
  const int E = in_sizes[1];                      // 1600000

  float* deg_out = (float*)d_ws;                  // [N] -> becomes inv_sqrt_out
  float* deg_in  = deg_out + N;                   // [N] -> becomes inv_sqrt_in
  float* agg     = (float*)d_out;                 // [N,64] accumulated in d_out, then GEMM'd in place

  const int B = 256;

  // 1) zero degree scratch + output accumulator
  zero_f32<<<(2 * N + B - 1) / B, B, 0, stream>>>(deg_out, 2 * N);
  zero_f32<<<(out_size + B - 1) / B, B, 0, stream>>>(agg, out_size);

  // 2) degree histograms (f32 atomics, L2-resident)
  degree_kernel<<<(E + B - 1) / B, B, 0, stream>>>(src, dst, deg_out, deg_in, E);

  // 3) inv-sqrt normalizers in place
  rsqrt_kernel<<<(2 * N + B - 1) / B, B, 0, stream>>>(deg_out, 2 * N);

  // 4) edge scatter: 16 lanes/edge * float4
  long long sthreads = (long long)E * 16;
  scatter_kernel<<<(int)((sthreads + B - 1) / B), B, 0, stream>>>(
      feature, src, dst, deg_out, deg_in, agg, E);

  // 5) WMMA GEMM + bias, in place on d_out
  int n_tiles = (N + 15) / 16;                    // 6250
  int waves_per_block = B / 32;                   // 8
  int gblocks = (n_tiles + waves_per_block - 1) / waves_per_block;
  gemm_wmma_kernel<<<gblocks, B, 0, stream>>>(agg, W2, b2, n_tiles);
}